// UnrolledSolver_21818433864390
// MI455X (gfx1250) — compile-verified
//
#include <hip/hip_runtime.h>
#include <hip/hip_bf16.h>

typedef __attribute__((ext_vector_type(2))) float v2f;
typedef __attribute__((ext_vector_type(8))) float v8f;

#define BGc 32
#define Nc 512
#define TWO_N 1024
#define EPGc 1024
#define Ec (BGc * EPGc)
#define T_STEPSc 5
#define EPS_REGc 1e-7f
#define VM_MINc 0.9f
#define VM_MAXc 1.1f
#define IMP_EPSc 1e-12f

// ---------------------------------------------------------------- zero
__global__ void pf_zero_kernel(float* __restrict__ J, float* __restrict__ Pacc,
                               float* __restrict__ Qacc) {
    size_t total = (size_t)BGc * TWO_N * TWO_N;
    size_t stride = (size_t)gridDim.x * blockDim.x;
    for (size_t i = (size_t)blockIdx.x * blockDim.x + threadIdx.x; i < total; i += stride)
        J[i] = 0.0f;
    int nn = BGc * Nc;
    for (int i = blockIdx.x * blockDim.x + threadIdx.x; i < nn;
         i += gridDim.x * blockDim.x) {
        Pacc[i] = 0.0f;
        Qacc[i] = 0.0f;
    }
}

// ---------------------------------------------------------------- edge scatter
__global__ void pf_edge_kernel(const float* __restrict__ x,
                               const float* __restrict__ br_r, const float* __restrict__ br_x,
                               const float* __restrict__ g_fr, const float* __restrict__ b_fr,
                               const float* __restrict__ g_to, const float* __restrict__ b_to,
                               const float* __restrict__ tap, const float* __restrict__ shift,
                               const int* __restrict__ src, const int* __restrict__ dst,
                               const int* __restrict__ bus_type,
                               float* __restrict__ J, float* __restrict__ Pacc,
                               float* __restrict__ Qacc) {
    int e = blockIdx.x * blockDim.x + threadIdx.x;
    if (e >= Ec) return;
    int b = e / EPGc;
    int s = src[e], d = dst[e];
    const float* xb = x + b * TWO_N;
    float va_s = xb[s], va_d = xb[d];
    float v_i = xb[Nc + s], v_j = xb[Nc + d];

    float rr = br_r[e] + IMP_EPSc;
    float bx = br_x[e];
    float den = rr * rr + bx * bx;
    float g_s = rr / den;
    float b_s = -bx / den;

    float ang = (va_s - va_d) - shift[e];
    float cf = cosf(ang), sf = sinf(ang);

    float tp = tap[e];
    float vij_tau = v_i * v_j / tp;
    float vi_tau = v_i / tp;
    float vi_tau_sq = vi_tau * vi_tau;
    float vj_tau = v_j / tp;

    float A_f = g_s * cf + b_s * sf;
    float B_f = g_s * sf - b_s * cf;
    float A_t = g_s * cf - b_s * sf;   // ct = cf, st = -sf
    float B_t = -g_s * sf - b_s * cf;

    float gsum_f = g_s + g_fr[e], bsum_f = b_s + b_fr[e];
    float gsum_t = g_s + g_to[e], bsum_t = b_s + b_to[e];

    float P_from = vi_tau_sq * gsum_f - vij_tau * A_f;
    float Q_from = -vi_tau_sq * bsum_f - vij_tau * B_f;
    float P_to = v_j * v_j * gsum_t - vij_tau * A_t;
    float Q_to = -v_j * v_j * bsum_t - vij_tau * B_t;

    atomicAdd(&Pacc[b * Nc + s], P_from);
    atomicAdd(&Pacc[b * Nc + d], P_to);
    atomicAdd(&Qacc[b * Nc + s], Q_from);
    atomicAdd(&Qacc[b * Nc + d], Q_to);

    float vi_c = fmaxf(v_i, 1e-12f);
    float dPfs_dth_s = vij_tau * B_f;
    float dPfs_dvm_s = 2.0f * vi_tau_sq / vi_c * gsum_f - vj_tau * A_f;
    float dPfs_dvm_d = -vi_tau * A_f;
    float dQfs_dth_s = -vij_tau * A_f;
    float dQfs_dvm_s = -2.0f * vi_tau_sq / vi_c * bsum_f - vj_tau * B_f;
    float dQfs_dvm_d = -vi_tau * B_f;
    float dPfd_dth_d = vij_tau * B_t;
    float dPfd_dvm_d = 2.0f * v_j * gsum_t - vi_tau * A_t;
    float dPfd_dvm_s = -vj_tau * A_t;
    float dQfd_dth_d = -vij_tau * A_t;
    float dQfd_dvm_d = -2.0f * v_j * bsum_t - vi_tau * B_t;
    float dQfd_dvm_s = -vj_tau * B_t;

    int bts = bus_type[b * Nc + s];
    int btd = bus_type[b * Nc + d];
    bool sl_s = (bts == 3), sl_d = (btd == 3);
    bool qm_s = (bts == 2) || sl_s;
    bool qm_d = (btd == 2) || sl_d;

    float* Jb = J + (size_t)b * TWO_N * TWO_N;
    // P-block rows (masked by slack)
    if (!sl_s) {
        atomicAdd(&Jb[(size_t)s * TWO_N + s], -dPfs_dth_s);
        atomicAdd(&Jb[(size_t)s * TWO_N + d], dPfs_dth_s);          // -dPfs_dth_d
        atomicAdd(&Jb[(size_t)s * TWO_N + Nc + s], -dPfs_dvm_s);
        atomicAdd(&Jb[(size_t)s * TWO_N + Nc + d], -dPfs_dvm_d);
    }
    if (!sl_d) {
        atomicAdd(&Jb[(size_t)d * TWO_N + d], -dPfd_dth_d);
        atomicAdd(&Jb[(size_t)d * TWO_N + s], dPfd_dth_d);          // -dPfd_dth_s
        atomicAdd(&Jb[(size_t)d * TWO_N + Nc + d], -dPfd_dvm_d);
        atomicAdd(&Jb[(size_t)d * TWO_N + Nc + s], -dPfd_dvm_s);
    }
    // Q-block rows (masked by pv|slack)
    if (!qm_s) {
        atomicAdd(&Jb[(size_t)(Nc + s) * TWO_N + s], -dQfs_dth_s);
        atomicAdd(&Jb[(size_t)(Nc + s) * TWO_N + d], dQfs_dth_s);
        atomicAdd(&Jb[(size_t)(Nc + s) * TWO_N + Nc + s], -dQfs_dvm_s);
        atomicAdd(&Jb[(size_t)(Nc + s) * TWO_N + Nc + d], -dQfs_dvm_d);
    }
    if (!qm_d) {
        atomicAdd(&Jb[(size_t)(Nc + d) * TWO_N + d], -dQfd_dth_d);
        atomicAdd(&Jb[(size_t)(Nc + d) * TWO_N + s], dQfd_dth_d);
        atomicAdd(&Jb[(size_t)(Nc + d) * TWO_N + Nc + d], -dQfd_dvm_d);
        atomicAdd(&Jb[(size_t)(Nc + d) * TWO_N + Nc + s], -dQfd_dvm_s);
    }
}

// ---------------------------------------------------------------- finalize F + diag
__global__ void pf_finalize_kernel(const float* __restrict__ x,
                                   const float* __restrict__ p_spec, const float* __restrict__ q_spec,
                                   const float* __restrict__ node_gs, const float* __restrict__ node_bs,
                                   const float* __restrict__ vm_sp, const int* __restrict__ bus_type,
                                   const float* __restrict__ Pacc, const float* __restrict__ Qacc,
                                   float* __restrict__ J, float* __restrict__ F) {
    int idx = blockIdx.x * blockDim.x + threadIdx.x;
    if (idx >= BGc * Nc) return;
    int b = idx / Nc, i = idx % Nc;
    float va = x[b * TWO_N + i];
    float vm = x[b * TWO_N + Nc + i];
    int bt = bus_type[idx];
    bool sl = (bt == 3);
    bool pvsl = (bt == 2) || sl;
    float gs = node_gs[idx], bs = node_bs[idx];
    float Pc = Pacc[idx] + vm * vm * gs;
    float Qc = Qacc[idx] - vm * vm * bs;
    F[b * TWO_N + i] = sl ? va : (p_spec[idx] - Pc);
    F[b * TWO_N + Nc + i] = pvsl ? (vm - vm_sp[idx]) : (q_spec[idx] - Qc);

    float* Jb = J + (size_t)b * TWO_N * TWO_N;
    if (sl)
        Jb[(size_t)i * TWO_N + i] += 1.0f;                 // ov on P diagonal
    else
        Jb[(size_t)i * TWO_N + Nc + i] += -2.0f * vm * gs; // dFP_dvm diag add
    if (pvsl)
        Jb[(size_t)(Nc + i) * TWO_N + Nc + i] += 1.0f;     // ov on Q diagonal
    else
        Jb[(size_t)(Nc + i) * TWO_N + Nc + i] += 2.0f * vm * bs;
    Jb[(size_t)i * TWO_N + i] += EPS_REGc;                 // eps * I
    Jb[(size_t)(Nc + i) * TWO_N + Nc + i] += EPS_REGc;
}

// ---------------------------------------------------------------- LU panel (width 16)
__global__ void __launch_bounds__(256) pf_panel_kernel(float* __restrict__ J, int k) {
    int b = blockIdx.x;
    float* A = J + (size_t)b * TWO_N * TWO_N;
    const int kc = k * 16;
    __shared__ float sInv;
    __shared__ float sRow[16];
    __shared__ float L16[16][17];

    // factor panel columns kc..kc+15 (no pivoting; eps-regularized system)
    for (int j = 0; j < 16; ++j) {
        int col = kc + j;
        if (threadIdx.x == 0) sInv = 1.0f / A[(size_t)col * TWO_N + col];
        if (threadIdx.x < 16) sRow[threadIdx.x] = A[(size_t)col * TWO_N + kc + threadIdx.x];
        __syncthreads();
        float ip = sInv;
        for (int i = col + 1 + threadIdx.x; i < TWO_N; i += blockDim.x) {
            float lij = A[(size_t)i * TWO_N + col] * ip;
            A[(size_t)i * TWO_N + col] = lij;
#pragma unroll
            for (int jj = 0; jj < 16; ++jj)
                if (jj > j) A[(size_t)i * TWO_N + kc + jj] -= lij * sRow[jj];
        }
        __syncthreads();
    }

    // load finished 16x16 L block (unit diag implicit)
    {
        int t = threadIdx.x;
        if (t < 256) L16[t >> 4][t & 15] = A[(size_t)(kc + (t >> 4)) * TWO_N + kc + (t & 15)];
    }
    __syncthreads();

    // U12: rows kc..kc+15, cols kc+16..TWO_N-1, each thread owns columns
    for (int c = kc + 16 + threadIdx.x; c < TWO_N; c += blockDim.x) {
        float u[16];
#pragma unroll
        for (int r = 0; r < 16; ++r) u[r] = A[(size_t)(kc + r) * TWO_N + c];
#pragma unroll
        for (int r = 1; r < 16; ++r) {
            float acc = u[r];
#pragma unroll
            for (int t = 0; t < 16; ++t)
                if (t < r) acc -= L16[r][t] * u[t];
            u[r] = acc;
        }
#pragma unroll
        for (int r = 1; r < 16; ++r) A[(size_t)(kc + r) * TWO_N + c] = u[r];
    }
}

// ---------------------------------------------------------------- trailing GEMM (WMMA f32)
// A22 -= L21 * U12 ; each wave owns a 32x32 macro tile (2x2 WMMA accumulators).
// The wave/tile indices are forced into SGPRs via readfirstlane so the edge-tile
// guards lower to scalar branches (EXEC stays all-ones around every WMMA).
__global__ void __launch_bounds__(256) pf_update_kernel(float* __restrict__ J, int k) {
    int b = blockIdx.x;
    float* A = J + (size_t)b * TWO_N * TWO_N;
    const int kc = k * 16;
    const int r0 = kc + 16;
    const int tm = (TWO_N - r0) >> 4;   // 16x16 tiles per dim
    const int tm2 = (tm + 1) >> 1;      // 32x32 macro tiles per dim
    int wave = __builtin_amdgcn_readfirstlane(
        (int)(blockIdx.y * (blockDim.x >> 5) + (threadIdx.x >> 5)));
    if (wave >= tm2 * tm2) return;      // scalar-uniform: EXEC untouched below
    int ti2 = wave / tm2, tj2 = wave % tm2;
    int lane = threadIdx.x & 31;
    int lh = lane >> 4;                 // lane half selects K pair / M+8
    int l16 = lane & 15;
    int row0 = r0 + ti2 * 32;
    int col0 = r0 + tj2 * 32;
    bool rhi = (ti2 * 2 + 1) < tm;      // scalar guards for odd tile counts
    bool chi = (tj2 * 2 + 1) < tm;

    v8f c00 = {}, c01 = {}, c10 = {}, c11 = {};
#pragma unroll
    for (int r = 0; r < 8; ++r) {
        int rr = row0 + r + lh * 8;     // C/D layout: lanes 16-31 hold M=r+8
        c00[r] = A[(size_t)rr * TWO_N + col0 + l16];
        if (chi) c01[r] = A[(size_t)rr * TWO_N + col0 + 16 + l16];
        if (rhi) c10[r] = A[(size_t)(rr + 16) * TWO_N + col0 + l16];
        if (rhi && chi) c11[r] = A[(size_t)(rr + 16) * TWO_N + col0 + 16 + l16];
    }
#pragma unroll
    for (int s = 0; s < 4; ++s) {
        int kb = kc + 4 * s + 2 * lh;   // A layout: lanes 0-15 K={0,1}, 16-31 K={2,3}
        v2f a0, a1, b0, b1;
        a0.x = -A[(size_t)(row0 + l16) * TWO_N + kb];       // negate L -> D = (-L)U + C
        a0.y = -A[(size_t)(row0 + l16) * TWO_N + kb + 1];
        if (rhi) {
            a1.x = -A[(size_t)(row0 + 16 + l16) * TWO_N + kb];
            a1.y = -A[(size_t)(row0 + 16 + l16) * TWO_N + kb + 1];
        } else { a1 = a0; }
        b0.x = A[(size_t)kb * TWO_N + col0 + l16];
        b0.y = A[(size_t)(kb + 1) * TWO_N + col0 + l16];
        if (chi) {
            b1.x = A[(size_t)kb * TWO_N + col0 + 16 + l16];
            b1.y = A[(size_t)(kb + 1) * TWO_N + col0 + 16 + l16];
        } else { b1 = b0; }
        c00 = __builtin_amdgcn_wmma_f32_16x16x4_f32(false, a0, false, b0, (short)0, c00, false, false);
        if (chi) c01 = __builtin_amdgcn_wmma_f32_16x16x4_f32(false, a0, false, b1, (short)0, c01, false, false);
        if (rhi) c10 = __builtin_amdgcn_wmma_f32_16x16x4_f32(false, a1, false, b0, (short)0, c10, false, false);
        if (rhi && chi) c11 = __builtin_amdgcn_wmma_f32_16x16x4_f32(false, a1, false, b1, (short)0, c11, false, false);
    }
#pragma unroll
    for (int r = 0; r < 8; ++r) {
        int rr = row0 + r + lh * 8;
        A[(size_t)rr * TWO_N + col0 + l16] = c00[r];
        if (chi) A[(size_t)rr * TWO_N + col0 + 16 + l16] = c01[r];
        if (rhi) A[(size_t)(rr + 16) * TWO_N + col0 + l16] = c10[r];
        if (rhi && chi) A[(size_t)(rr + 16) * TWO_N + col0 + 16 + l16] = c11[r];
    }
}

// ---------------------------------------------------------------- triangular solves + x update
__global__ void __launch_bounds__(256) pf_solve_kernel(const float* __restrict__ J,
                                                       float* __restrict__ F,
                                                       float* __restrict__ x) {
    int b = blockIdx.x;
    const float* A = J + (size_t)b * TWO_N * TWO_N;
    float* f = F + b * TWO_N;
    // forward substitution with unit-lower L
    for (int c = 0; c < TWO_N - 1; ++c) {
        __syncthreads();
        float fc = f[c];
        for (int i = c + 1 + threadIdx.x; i < TWO_N; i += blockDim.x)
            f[i] -= A[(size_t)i * TWO_N + c] * fc;
        __syncthreads();
    }
    // backward substitution with U
    for (int c = TWO_N - 1; c >= 0; --c) {
        __syncthreads();
        if (threadIdx.x == 0) f[c] = f[c] / A[(size_t)c * TWO_N + c];
        __syncthreads();
        float fc = f[c];
        for (int i = threadIdx.x; i < c; i += blockDim.x)
            f[i] -= A[(size_t)i * TWO_N + c] * fc;
    }
    __syncthreads();
    // x_new = x - dx, clip vm half
    for (int j = threadIdx.x; j < TWO_N; j += blockDim.x) {
        float xn = x[b * TWO_N + j] - f[j];
        if (j >= Nc) xn = fminf(fmaxf(xn, VM_MINc), VM_MAXc);
        x[b * TWO_N + j] = xn;
    }
}

// ---------------------------------------------------------------- launch
extern "C" void kernel_launch(void* const* d_in, const int* in_sizes, int n_in,
                              void* d_out, int out_size, void* d_ws, size_t ws_size,
                              hipStream_t stream) {
    const float* x0      = (const float*)d_in[0];
    const float* br_r    = (const float*)d_in[1];
    const float* br_x    = (const float*)d_in[2];
    const float* g_fr    = (const float*)d_in[3];
    const float* b_fr    = (const float*)d_in[4];
    const float* g_to    = (const float*)d_in[5];
    const float* b_to    = (const float*)d_in[6];
    const float* tap     = (const float*)d_in[7];
    const float* shift   = (const float*)d_in[8];
    const float* p_spec  = (const float*)d_in[9];
    const float* q_spec  = (const float*)d_in[10];
    const float* node_gs = (const float*)d_in[11];
    const float* node_bs = (const float*)d_in[12];
    const float* vm_sp   = (const float*)d_in[13];
    const int*   src     = (const int*)d_in[14];
    const int*   dst     = (const int*)d_in[15];
    const int*   bus_type= (const int*)d_in[17];   // be (16) recomputed as e/EPG

    float* x = (float*)d_out;                       // live state lives in d_out

    size_t jFloats = (size_t)BGc * TWO_N * TWO_N;
    size_t need = (jFloats + 2 * (size_t)BGc * Nc + (size_t)BGc * TWO_N) * sizeof(float);
    if (ws_size < need) return;
    float* J    = (float*)d_ws;
    float* Pacc = J + jFloats;
    float* Qacc = Pacc + (size_t)BGc * Nc;
    float* F    = Qacc + (size_t)BGc * Nc;

    hipMemcpyAsync(x, x0, (size_t)BGc * TWO_N * sizeof(float),
                   hipMemcpyDeviceToDevice, stream);

    for (int t = 0; t < T_STEPSc; ++t) {
        pf_zero_kernel<<<4096, 256, 0, stream>>>(J, Pacc, Qacc);
        pf_edge_kernel<<<Ec / 256, 256, 0, stream>>>(x, br_r, br_x, g_fr, b_fr, g_to, b_to,
                                                     tap, shift, src, dst, bus_type,
                                                     J, Pacc, Qacc);
        pf_finalize_kernel<<<(BGc * Nc) / 256, 256, 0, stream>>>(x, p_spec, q_spec,
                                                                 node_gs, node_bs, vm_sp,
                                                                 bus_type, Pacc, Qacc, J, F);
        for (int k = 0; k < TWO_N / 16; ++k) {
            pf_panel_kernel<<<BGc, 256, 0, stream>>>(J, k);
            int tm = (TWO_N - (k + 1) * 16) / 16;
            if (tm > 0) {
                int tm2 = (tm + 1) / 2;
                int wavesNeeded = tm2 * tm2;
                int by = (wavesNeeded + 7) / 8;   // 8 waves per 256-thread block
                dim3 grid(BGc, by);
                pf_update_kernel<<<grid, 256, 0, stream>>>(J, k);
            }
        }
        pf_solve_kernel<<<BGc, 256, 0, stream>>>(J, F, x);
    }
}